// RHN_90735479095761
// MI455X (gfx1250) — compile-verified
//
#include <hip/hip_runtime.h>
#include <math.h>

// Problem constants (from reference)
#define TT   2048
#define BB   32
#define DD   256
#define HH   256
#define W3H  768     // 3*H, row stride of the weight matrices
#define NPRE 512     // only h_pre and t_pre columns are live (c_pre is dead)
#define HSTR 260     // LDS row stride for h (bank-conflict-free A-frag loads)

typedef __attribute__((ext_vector_type(2))) float v2f;
typedef __attribute__((ext_vector_type(8))) float v8f;

// preX is stored in WMMA C-fragment layout so the recurrence loads it with
// two b128s per M-tile:  flat = (((t*32 + w)*2 + mt)*32 + lane)*8 + r
//   w  = column tile (0..31) == recurrence wave id
//   mt = M-tile (batch rows 0-15 / 16-31)
//   lane/r = WMMA C fragment coordinates
__device__ __forceinline__ size_t px_off(int t, int w, int mt, int lane) {
  return ((((size_t)(t * 32 + w) * 2 + mt) * 32 + lane) * 8);
}

// ---------------------------------------------------------------------------
// Kernel 1: time-parallel input GEMM.
//   preX[frag layout] = X[M=T*B, 256] @ W[:, 0:512] + bias[0:512]
// f32 WMMA 16x16x4. Block = 256 thr (8 waves): 128 rows x 128 cols per block.
// B-fragments staged swizzled in LDS (128 KB) so each frag is one ds b64 read.
// ---------------------------------------------------------------------------
__global__ __launch_bounds__(256) void rhn_gemm_x(
    const float* __restrict__ X, const float* __restrict__ W,
    const float* __restrict__ bias, float* __restrict__ preX) {
  extern __shared__ char smem[];
  v2f* wtile = (v2f*)smem;                 // [64 kchunks][8 ntiles][32 lanes]
  const int tid  = threadIdx.x;
  const int lane = tid & 31;
  const int wave = tid >> 5;
  const int m0   = blockIdx.x * 128 + wave * 16;
  const int n0   = blockIdx.y * 128;

  // Stage weight tile in WMMA B-fragment layout: lane L -> (K=(L/16)*2+v, N=L%16)
  for (int e = tid; e < 64 * 8 * 32; e += 256) {
    const int c  = e >> 8;                 // kchunk
    const int nt = (e >> 5) & 7;           // ntile within block
    const int l  = e & 31;                 // lane
    const int k  = 4 * c + ((l >> 4) << 1);
    const int n  = n0 + nt * 16 + (l & 15);
    v2f wv;
    wv.x = W[k * W3H + n];
    wv.y = W[(k + 1) * W3H + n];
    wtile[e] = wv;
  }
  __syncthreads();

  // Accumulators initialized with bias (bias depends on N only == lane%16)
  v8f acc[8];
#pragma unroll
  for (int nt = 0; nt < 8; ++nt) {
    const float bv = bias[n0 + nt * 16 + (lane & 15)];
#pragma unroll
    for (int r = 0; r < 8; ++r) acc[nt][r] = bv;
  }

  // A-fragment: lane L -> (M=L%16, K=4c+(L/16)*2 + v); one global b64 per chunk
  const int arow = m0 + (lane & 15);
  const float* aptr = X + (size_t)arow * DD + ((lane >> 4) << 1);

#pragma unroll 8
  for (int c = 0; c < 64; ++c) {
    const v2f a = *(const v2f*)(aptr + 4 * c);
#pragma unroll
    for (int nt = 0; nt < 8; ++nt) {
      const v2f b = wtile[(c * 8 + nt) * 32 + lane];
      acc[nt] = __builtin_amdgcn_wmma_f32_16x16x4_f32(
          false, a, false, b, (short)0, acc[nt], false, false);
    }
  }

  // Store C-fragments directly in the recurrence's load layout (2x b128 each)
  const int tIdx = m0 >> 5;                // time index of this 16-row tile
  const int mt   = (m0 >> 4) & 1;          // which half of the batch
#pragma unroll
  for (int nt = 0; nt < 8; ++nt) {
    const int w = (n0 >> 4) + nt;          // recurrence wave owning these cols
    *(v8f*)(preX + px_off(tIdx, w, mt, lane)) = acc[nt];
  }
}

__device__ __forceinline__ float sigf(float x) {
  return 1.0f / (1.0f + __expf(-x));
}

// ---------------------------------------------------------------------------
// Kernel 2: the sequential recurrence. ONE workgroup, 1024 threads = 32 waves.
//   h stays in LDS; Whh[:, 0:512] stays in VGPRs (64 v2f B-frags per wave).
//   Wave w owns pre-activation columns [16w, 16w+16): waves 0-15 = h_pre,
//   waves 16-31 = t_pre (lane-for-lane matched fragment coordinates).
// Per step: 128 chained v_wmma_f32_16x16x4_f32 per wave; preX for step t+1 is
// prefetched into registers underneath the WMMA chain (software pipeline).
// ---------------------------------------------------------------------------
__global__ __launch_bounds__(1024) void rhn_recur(
    const float* __restrict__ preX, const float* __restrict__ Whh,
    const int* __restrict__ length, float* __restrict__ out,
    float* __restrict__ hn) {
  extern __shared__ char smem[];
  float* hbuf = (float*)smem;                       // [32][HSTR]
  float* tbuf = hbuf + 32 * HSTR;                   // [16][2][32 lanes][8]
  int*   lbuf = (int*)(tbuf + 16 * 2 * 32 * 8);     // [32]

  const int tid  = threadIdx.x;
  const int lane = tid & 31;
  const int wave = tid >> 5;                        // 0..31

  // h0 = 0; cache lengths
  for (int i = tid; i < 32 * HSTR; i += 1024) hbuf[i] = 0.0f;
  if (tid < 32) lbuf[tid] = length[tid];
  __syncthreads();

  // Preload this wave's Whh column tile into registers as B-fragments:
  // lane L -> (K = 4c + (L/16)*2 + v, N = 16*wave + L%16)
  const int ncol = wave * 16 + (lane & 15);         // 0..511
  const int ksub = (lane >> 4) << 1;
  v2f wreg[64];
#pragma unroll
  for (int c = 0; c < 64; ++c) {
    const int k = 4 * c + ksub;
    v2f wv;
    wv.x = Whh[k * W3H + ncol];
    wv.y = Whh[(k + 1) * W3H + ncol];
    wreg[c] = wv;
  }

  const int crow = (lane >> 4) << 3;                // C-frag row offset: 0 or 8
  const int am   = lane & 15;                       // A-frag row = M = lane%16
  const bool isH = (wave < 16);
  const int  w16 = isH ? wave : (wave - 16);

  // Software pipeline: cur holds preX fragments for the step about to run.
  v8f cur0 = *(const v8f*)(preX + px_off(0, wave, 0, lane));
  v8f cur1 = *(const v8f*)(preX + px_off(0, wave, 1, lane));

  for (int t = 0; t < TT; ++t) {
    v8f acc0 = cur0;
    v8f acc1 = cur1;

    // Prefetch next step's fragments; hidden under the 128-WMMA chain below.
    const int tn = (t + 1) & (TT - 1);              // harmless wrap at t=2047
    cur0 = *(const v8f*)(preX + px_off(tn, wave, 0, lane));
    cur1 = *(const v8f*)(preX + px_off(tn, wave, 1, lane));

    // pre += h @ Whh  (K = 256 -> 64 chained f32 WMMAs per M-tile)
#pragma unroll
    for (int c = 0; c < 64; ++c) {
      const v2f a0 = *(const v2f*)(hbuf + am * HSTR + 4 * c + ksub);
      const v2f a1 = *(const v2f*)(hbuf + (16 + am) * HSTR + 4 * c + ksub);
      acc0 = __builtin_amdgcn_wmma_f32_16x16x4_f32(
          false, a0, false, wreg[c], (short)0, acc0, false, false);
      acc1 = __builtin_amdgcn_wmma_f32_16x16x4_f32(
          false, a1, false, wreg[c], (short)0, acc1, false, false);
    }

    if (!isH) {  // t_pre waves publish their fragments (lane-matched layout)
      float* tb0 = tbuf + ((w16 * 2 + 0) * 32 + lane) * 8;
      float* tb1 = tbuf + ((w16 * 2 + 1) * 32 + lane) * 8;
#pragma unroll
      for (int r = 0; r < 8; ++r) { tb0[r] = acc0[r]; tb1[r] = acc1[r]; }
    }
    __syncthreads();

    if (isH) {   // fuse gates, mask, h update, out[t] store
#pragma unroll
      for (int mt = 0; mt < 2; ++mt) {
        const float* tb = tbuf + ((w16 * 2 + mt) * 32 + lane) * 8;
        const v8f hp = mt ? acc1 : acc0;
#pragma unroll
        for (int r = 0; r < 8; ++r) {
          const int b   = mt * 16 + r + crow;       // batch row
          const int col = w16 * 16 + (lane & 15);   // hidden column
          const float hold = hbuf[b * HSTR + col];
          const float tg = sigf(tb[r]);
          const float cg = sigf(tg);                // bug-faithful sigmoid(sigmoid)
          const float s  = tanhf(hp[r]) * tg + hold * cg;
          const float hv = (t < lbuf[b]) ? s : hold;
          hbuf[b * HSTR + col] = hv;
          out[(size_t)(t * BB + b) * HH + col] = hv;
        }
      }
    }
    __syncthreads();  // h fully updated before next step's A-frag reads
  }

  // final hidden state
  for (int i = tid; i < BB * HH; i += 1024)
    hn[i] = hbuf[(i >> 8) * HSTR + (i & 255)];
}

// ---------------------------------------------------------------------------
// kernel_launch: preX(GEMM) -> recurrence, twice (layer 1 input = layer 0 out,
// which is staged in d_out and then overwritten by layer-1 recurrence).
// Workspace use: T*B*512 f32 = 128 MB in d_ws.
// ---------------------------------------------------------------------------
extern "C" void kernel_launch(void* const* d_in, const int* in_sizes, int n_in,
                              void* d_out, int out_size, void* d_ws,
                              size_t ws_size, hipStream_t stream) {
  const float* input_ = (const float*)d_in[0];
  const int*   length = (const int*)d_in[1];
  const float* Wih0   = (const float*)d_in[2];
  const float* Whh0   = (const float*)d_in[3];
  const float* b0     = (const float*)d_in[4];
  const float* Wih1   = (const float*)d_in[5];
  const float* Whh1   = (const float*)d_in[6];
  const float* b1     = (const float*)d_in[7];

  float* out  = (float*)d_out;                       // [T,B,H]
  float* hn   = out + (size_t)TT * BB * HH;          // [2,B,H]
  float* preX = (float*)d_ws;                        // 128 MB, frag layout

  const dim3  g1(512, 4);                            // 128-row x 128-col tiles
  const size_t lds1 = (size_t)64 * 8 * 32 * sizeof(v2f);           // 128 KB
  const size_t lds2 = (size_t)(32 * HSTR + 16 * 2 * 32 * 8 + 32) * 4; // ~65 KB

  // Layer 0
  rhn_gemm_x<<<g1, 256, lds1, stream>>>(input_, Wih0, b0, preX);
  rhn_recur<<<dim3(1), 1024, lds2, stream>>>(preX, Whh0, length, out, hn);
  // Layer 1 (input = out0 staged in d_out; preX fully precomputed before
  // the recurrence overwrites d_out with out1)
  rhn_gemm_x<<<g1, 256, lds1, stream>>>(out, Wih1, b1, preX);
  rhn_recur<<<dim3(1), 1024, lds2, stream>>>(preX, Whh1, length, out,
                                             hn + BB * HH);
}